// PlaneSweepCostVolume_5317169513238
// MI455X (gfx1250) — compile-verified
//
#include <hip/hip_runtime.h>
#include <stdint.h>

#define Vn 4
#define Cn 32
#define Hn 128
#define Wn 160
#define Dn 32
#define HWn (Hn*Wn)          // 20480
#define TILE 256
#define NTILES (HWn/TILE)    // 80

typedef uint32_t u32x4 __attribute__((ext_vector_type(4)));
typedef uint32_t u32x8 __attribute__((ext_vector_type(8)));

__global__ __launch_bounds__(TILE) void psv_kernel(
    const float* __restrict__ ref_feat,
    const float* __restrict__ src_feats,
    const float* __restrict__ K_ref,
    const float* __restrict__ c2w_ref,
    const float* __restrict__ K_srcs,
    const float* __restrict__ c2w_srcs,
    float* __restrict__ out)
{
  extern __shared__ float smem[];     // [0, Cn*TILE) = ref tile, then per-view mats
  float* tile = smem;                 // tile[c*TILE + t]
  float* mats = smem + Cn*TILE;       // per view: M[9], t[3], fx,fy,cx,cy (16 floats)

  const int t = threadIdx.x;
  const int pix0 = (int)blockIdx.x * TILE;
  const int d = (int)blockIdx.y;
  const int n = pix0 + t;

  // ---------- wave 0: TDM async tensor load of ref tile (Cn x TILE fp32) ----------
  if (t < 32) {
    uint32_t lds_off = (uint32_t)(uintptr_t)tile;                 // flat LDS ptr low 32b = LDS byte offset
    uint64_t ga = (uint64_t)(uintptr_t)(ref_feat + pix0);         // tile start in tensor
    u32x4 g0;
    g0[0] = 1u;                                                   // count=1, no gather mode
    g0[1] = lds_off;                                              // lds_addr
    g0[2] = (uint32_t)ga;                                         // global_addr[31:0]
    g0[3] = ((uint32_t)(ga >> 32) & 0x01FFFFFFu) | (2u << 30);    // global_addr[56:32] | type=2
    u32x8 g1;
    g1[0] = (2u << 16);                                           // data_size = 4B, wg_mask=0
    g1[1] = ((uint32_t)HWn) << 16;                                // tensor_dim0[15:0] @ [63:48]
    g1[2] = ((uint32_t)Cn) << 16;                                 // tensor_dim1[15:0] @ [95:80] (dim0 hi16 = 0)
    g1[3] = ((uint32_t)TILE) << 16;                               // tile_dim0 @ [127:112]
    g1[4] = (uint32_t)Cn;                                         // tile_dim1 @ [143:128], tile_dim2 = 0
    g1[5] = (uint32_t)HWn;                                        // tensor_dim0_stride[31:0] (elements)
    g1[6] = 0u;                                                   // stride0 hi16 / stride1 lo16
    g1[7] = 0u;
    asm volatile("tensor_load_to_lds %0, %1" :: "s"(g0), "s"(g1) : "memory");
    __builtin_amdgcn_s_wait_tensorcnt(0);
  }

  // ---------- thread 0: fold camera algebra into per-view affine maps ----------
  if (t == 0) {
    float a00=K_ref[0], a01=K_ref[1], a02=K_ref[2];
    float a10=K_ref[3], a11=K_ref[4], a12=K_ref[5];
    float a20=K_ref[6], a21=K_ref[7], a22=K_ref[8];
    float det = a00*(a11*a22-a12*a21) - a01*(a10*a22-a12*a20) + a02*(a10*a21-a11*a20);
    float id = 1.0f/det;
    float ki[9] = {
      (a11*a22-a12*a21)*id, (a02*a21-a01*a22)*id, (a01*a12-a02*a11)*id,
      (a12*a20-a10*a22)*id, (a00*a22-a02*a20)*id, (a02*a10-a00*a12)*id,
      (a10*a21-a11*a20)*id, (a01*a20-a00*a21)*id, (a00*a11-a01*a10)*id };
    float Rr[9], tr[3];
    for (int i=0;i<3;i++){ for(int j=0;j<3;j++) Rr[i*3+j]=c2w_ref[i*4+j]; tr[i]=c2w_ref[i*4+3]; }
    float A[9];
    for (int i=0;i<3;i++) for (int j=0;j<3;j++)
      A[i*3+j] = Rr[i*3+0]*ki[j] + Rr[i*3+1]*ki[3+j] + Rr[i*3+2]*ki[6+j];
    for (int v=0; v<Vn; v++) {
      const float* c2w = c2w_srcs + v*16;
      float Rc[9], tc[3];
      for (int i=0;i<3;i++){ for(int j=0;j<3;j++) Rc[i*3+j]=c2w[i*4+j]; tc[i]=c2w[i*4+3]; }
      float Rw[9], tw[3];
      for (int i=0;i<3;i++) for(int j=0;j<3;j++) Rw[i*3+j]=Rc[j*3+i];    // rigid inverse
      for (int i=0;i<3;i++) tw[i] = -(Rw[i*3+0]*tc[0]+Rw[i*3+1]*tc[1]+Rw[i*3+2]*tc[2]);
      float* m = mats + v*16;
      for (int i=0;i<3;i++) for (int j=0;j<3;j++)
        m[i*3+j] = Rw[i*3+0]*A[j] + Rw[i*3+1]*A[3+j] + Rw[i*3+2]*A[6+j]; // M = Rw2c * Rc2w_ref * Kinv
      for (int i=0;i<3;i++)
        m[9+i] = Rw[i*3+0]*tr[0]+Rw[i*3+1]*tr[1]+Rw[i*3+2]*tr[2] + tw[i];
      const float* Ks = K_srcs + v*9;
      m[12]=Ks[0]; m[13]=Ks[4]; m[14]=Ks[2]; m[15]=Ks[5];               // fx fy cx cy
    }
  }
  __syncthreads();

  // ---------- per-thread: one pixel at one depth plane ----------
  const float inv0 = 1.0f/15.0f, inv1 = 1.0f/0.5f;
  float invd  = inv0 + (inv1 - inv0) * ((float)d * (1.0f/(float)(Dn-1)));
  float depth = 1.0f / invd;

  float px = (float)(n % Wn);
  float py = (float)(n / Wn);

  // normalized ref feature (from LDS tile, held in VGPRs)
  float refn[Cn];
  float ss = 0.0f;
  #pragma unroll
  for (int c=0;c<Cn;c++){ float f = tile[c*TILE + t]; refn[c]=f; ss = fmaf(f,f,ss); }
  float rinv = 1.0f / fmaxf(sqrtf(ss), 1e-12f);
  #pragma unroll
  for (int c=0;c<Cn;c++) refn[c] *= rinv;

  float acc = 0.0f;
  #pragma unroll
  for (int v=0; v<Vn; v++) {
    const float* m = mats + v*16;
    float rx = m[0]*px + m[1]*py + m[2];
    float ry = m[3]*px + m[4]*py + m[5];
    float rz = m[6]*px + m[7]*py + m[8];
    float X  = fmaf(depth, rx, m[9]);
    float Y  = fmaf(depth, ry, m[10]);
    float Zr = fmaf(depth, rz, m[11]);
    float Z  = fmaxf(Zr, 1e-6f);
    float u  = fmaf(m[12], X/Z, m[14]);
    float vv = fmaf(m[13], Y/Z, m[15]);
    float gx = 2.0f*(u *(1.0f/(float)(Wn-1))) - 1.0f;
    float gy = 2.0f*(vv*(1.0f/(float)(Hn-1))) - 1.0f;
    float valid = (gx>=-1.0f && gx<=1.0f && gy>=-1.0f && gy<=1.0f && Zr>1e-6f) ? 1.0f : 0.0f;

    float x0 = floorf(u), y0 = floorf(vv);
    float wx1 = u - x0, wy1 = vv - y0;
    int x0i=(int)x0, y0i=(int)y0, x1i=x0i+1, y1i=y0i+1;
    float m00 = (x0i>=0 && x0i<Wn && y0i>=0 && y0i<Hn) ? 1.0f : 0.0f;
    float m10 = (x1i>=0 && x1i<Wn && y0i>=0 && y0i<Hn) ? 1.0f : 0.0f;
    float m01 = (x0i>=0 && x0i<Wn && y1i>=0 && y1i<Hn) ? 1.0f : 0.0f;
    float m11 = (x1i>=0 && x1i<Wn && y1i>=0 && y1i<Hn) ? 1.0f : 0.0f;
    int cx0 = min(max(x0i,0),Wn-1), cx1 = min(max(x1i,0),Wn-1);
    int cy0 = min(max(y0i,0),Hn-1), cy1 = min(max(y1i,0),Hn-1);
    int i00 = cy0*Wn + cx0, i10 = cy0*Wn + cx1;
    int i01 = cy1*Wn + cx0, i11 = cy1*Wn + cx1;
    float w00 = (1.0f-wx1)*(1.0f-wy1)*m00;
    float w10 = wx1*(1.0f-wy1)*m10;
    float w01 = (1.0f-wx1)*wy1*m01;
    float w11 = wx1*wy1*m11;

    const float* base = src_feats + v*(Cn*HWn);
    float dotp = 0.0f, sq = 0.0f;
    #pragma unroll
    for (int c=0;c<Cn;c++) {
      const float* bc = base + c*HWn;
      float f00=bc[i00], f10=bc[i10], f01=bc[i01], f11=bc[i11];
      float wv = fmaf(w00,f00, fmaf(w10,f10, fmaf(w01,f01, w11*f11)));
      sq   = fmaf(wv, wv, sq);
      dotp = fmaf(refn[c], wv, dotp);
    }
    float nrm = fmaxf(sqrtf(sq), 1e-12f);
    acc += (dotp / nrm) * valid;
  }

  out[d*HWn + n] = acc * (1.0f/(float)Vn);       // cost volume
  out[Dn*HWn + d*HWn + n] = depth;               // depth planes
}

extern "C" void kernel_launch(void* const* d_in, const int* in_sizes, int n_in,
                              void* d_out, int out_size, void* d_ws, size_t ws_size,
                              hipStream_t stream) {
  const float* ref_feat  = (const float*)d_in[0];
  const float* src_feats = (const float*)d_in[1];
  const float* K_ref     = (const float*)d_in[2];
  const float* c2w_ref   = (const float*)d_in[3];
  const float* K_srcs    = (const float*)d_in[4];
  const float* c2w_srcs  = (const float*)d_in[5];
  float* out = (float*)d_out;

  dim3 grid(NTILES, Dn);
  size_t shmem = (Cn*TILE + Vn*16) * sizeof(float);   // 32 KB tile + mats
  hipLaunchKernelGGL(psv_kernel, grid, dim3(TILE), shmem, stream,
                     ref_feat, src_feats, K_ref, c2w_ref, K_srcs, c2w_srcs, out);
}